// MultiscaleDeformableAttention_60722247630994
// MI455X (gfx1250) — compile-verified
//
#include <hip/hip_runtime.h>
#include <math.h>

typedef __attribute__((ext_vector_type(2))) float v2f;
typedef __attribute__((ext_vector_type(8))) float v8f;

constexpr int kEMB = 256;
constexpr int kH   = 8;
constexpr int kL   = 4;
constexpr int kP   = 4;
constexpr int kCp  = 32;              // HID / H
constexpr int kI   = 11253;           // 92^2 + 46^2 + 23^2 + 12^2
constexpr int kB   = 2;
constexpr int kN   = 11253;           // N == I in this problem
constexpr int kM   = kB * kN;         // 22506 rows for every GEMM

// ---------------------------------------------------------------------------
// C[M,N] = A[M,K] @ W[K,N] + bias[N], all f32 row-major.
// One wave32 computes a 16x64 strip: 4 16x16 WMMA tiles sharing one A fragment.
// N, K are compile-time so all inner-loop addresses are base + imm24 offsets;
// pointer-bump once per 8-step chunk. V_WMMA_F32_16X16X4_F32 layouts:
//   A 16x4 : lane<16 holds row M=lane, K pair {k,k+1}; lane>=16 -> {k+2,k+3}
//   B 4x16 : lane&15 = column, lane half selects the same K pair
//   C/D    : vgpr r -> row r + 8*(lane>=16), col = lane&15
// ---------------------------------------------------------------------------
template <int N, int K>
__global__ __launch_bounds__(256)
void gemm_bias_wmma_f32(const float* __restrict__ A, const float* __restrict__ W,
                        const float* __restrict__ bias, float* __restrict__ C,
                        int M) {
  constexpr int CT = 4;                         // column tiles per wave
  static_assert(N % (16 * CT) == 0 && K % 32 == 0, "shape");
  const int lane = threadIdx.x & 31;
  const int wave = blockIdx.x * 8 + (threadIdx.x >> 5);
  const int groupsN = N / (16 * CT);
  const int tilesM  = (M + 15) >> 4;
  if (wave >= tilesM * groupsN) return;         // wave-uniform early-out
  const int tm = wave / groupsN;
  const int tg = wave - tm * groupsN;
  const int row0 = tm << 4;
  const int col0 = tg * (16 * CT);
  const int half = lane >> 4;                   // K-pair select
  const int l16  = lane & 15;

  int arow = row0 + l16;
  if (arow >= M) arow = M - 1;                  // clamp loads on edge tile
  const float* pA = A + arow * K + 2 * half;    // A: single-use -> NT loads
  const float* pB = W + (2 * half) * N + col0 + l16;

  v8f acc[CT] = {};
  for (int kc = 0; kc < K; kc += 32) {
    #pragma unroll
    for (int ks = 0; ks < 32; ks += 4) {        // all offsets are immediates
      const v2f a = __builtin_nontemporal_load(
          reinterpret_cast<const v2f*>(pA + ks));
      v2f b[CT];
      #pragma unroll
      for (int t = 0; t < CT; ++t) {
        b[t].x = pB[ks * N + t * 16];
        b[t].y = pB[(ks + 1) * N + t * 16];
      }
      #pragma unroll
      for (int t = 0; t < CT; ++t)
        acc[t] = __builtin_amdgcn_wmma_f32_16x16x4_f32(
            false, a, false, b[t], (short)0, acc[t], false, false);
    }
    pA += 32;
    pB += 32 * N;
  }

  #pragma unroll
  for (int t = 0; t < CT; ++t) {
    const int col = col0 + t * 16 + l16;
    const float bv = bias[col];
    #pragma unroll
    for (int r = 0; r < 8; ++r) {
      const int row = row0 + r + 8 * half;
      if (row < M) C[row * N + col] = acc[t][r] + bv;
    }
  }
}

// ---------------------------------------------------------------------------
// Softmax over the 16 attention logits per (b,n,h) and sampling-point prep,
// in place on qp laid out (B,N,H,L,P,3): [...,0]+=ref_x, [...,1]+=ref_y,
// [...,2] = softmax over the L*P logits.
// ---------------------------------------------------------------------------
__global__ __launch_bounds__(256)
void softmax_prep(float* __restrict__ qp, const float* __restrict__ ref) {
  const int i = blockIdx.x * blockDim.x + threadIdx.x;   // (b*N+n)*H + h
  if (i >= kM * kH) return;
  const int bn = i >> 3;
  float* q = qp + i * (kL * kP * 3);
  const float rx = ref[bn * 2 + 0];
  const float ry = ref[bn * 2 + 1];

  float e[16];
  float mx = -3.4e38f;
  #pragma unroll
  for (int j = 0; j < 16; ++j) { e[j] = q[3 * j + 2]; mx = fmaxf(mx, e[j]); }
  float s = 0.f;
  #pragma unroll
  for (int j = 0; j < 16; ++j) { e[j] = __expf(e[j] - mx); s += e[j]; }
  const float inv = 1.f / s;
  #pragma unroll
  for (int j = 0; j < 16; ++j) {
    q[3 * j + 0] += rx;
    q[3 * j + 1] += ry;
    q[3 * j + 2]  = e[j] * inv;
  }
}

// ---------------------------------------------------------------------------
// Deformable bilinear sampling. One wave32 per (b,n,h); lane = channel (Cp=32).
// img_p layout (B,I,H,Cp) -> consecutive lanes read consecutive addresses:
// each corner gather is one coalesced 128B transaction, served out of L2
// (img_p = 23 MB << 192 MB L2).
// ---------------------------------------------------------------------------
__global__ __launch_bounds__(256)
void msda_sample(const float* __restrict__ imgp, const float* __restrict__ qp,
                 float* __restrict__ msda) {
  const int lane = threadIdx.x & 31;
  const int wid  = blockIdx.x * 8 + (threadIdx.x >> 5); // (b*N+n)*H + h
  if (wid >= kM * kH) return;
  const int h  = wid & 7;
  const int bn = wid >> 3;
  const int b  = bn / kN;
  const float* q = qp + wid * (kL * kP * 3);

  const int HW[4]    = {92, 46, 23, 12};
  const int START[4] = {0, 8464, 10580, 11109};
  const int S = kH * kCp;                               // 256 floats / spatial pos

  float acc = 0.f;
  #pragma unroll
  for (int l = 0; l < 4; ++l) {
    const int w = HW[l];
    const float wm1 = (float)(w - 1);
    const int base = ((b * kI + START[l]) * kH + h) * kCp + lane;
    #pragma unroll
    for (int p = 0; p < 4; ++p) {
      const int j = l * 4 + p;
      const float sx = q[3 * j + 0], sy = q[3 * j + 1], awt = q[3 * j + 2];
      const float x = fminf(fmaxf(sx * wm1, 0.f), wm1);
      const float y = fminf(fmaxf(sy * wm1, 0.f), wm1);
      const float x0f = floorf(x), y0f = floorf(y);
      const float wx = x - x0f, wy = y - y0f;
      const int x0 = (int)x0f, y0 = (int)y0f;
      const int x1 = min(x0 + 1, w - 1), y1 = min(y0 + 1, w - 1);
      const float v00 = imgp[base + (y0 * w + x0) * S];
      const float v01 = imgp[base + (y0 * w + x1) * S];
      const float v10 = imgp[base + (y1 * w + x0) * S];
      const float v11 = imgp[base + (y1 * w + x1) * S];
      const float vtop = v00 + wx * (v01 - v00);
      const float vbot = v10 + wx * (v11 - v10);
      acc += awt * (vtop + wy * (vbot - vtop));
    }
  }
  msda[bn * (kH * kCp) + h * kCp + lane] = acc;
}

// ---------------------------------------------------------------------------
extern "C" void kernel_launch(void* const* d_in, const int* in_sizes, int n_in,
                              void* d_out, int out_size, void* d_ws, size_t ws_size,
                              hipStream_t stream) {
  (void)in_sizes; (void)n_in; (void)out_size; (void)ws_size;
  const float* img     = (const float*)d_in[0];
  // d_in[1] = shapes (int32) — compile-time constant here
  const float* queries = (const float*)d_in[2];
  const float* refpts  = (const float*)d_in[3];
  const float* W_img   = (const float*)d_in[4];
  const float* b_img   = (const float*)d_in[5];
  const float* W_q     = (const float*)d_in[6];
  const float* b_q     = (const float*)d_in[7];
  const float* W_out   = (const float*)d_in[8];
  const float* b_out   = (const float*)d_in[9];
  float* out  = (float*)d_out;

  float* ws   = (float*)d_ws;
  float* imgp = ws;                          // kM*256 floats  (23.0 MB)
  float* qp   = imgp + (size_t)kM * 256;     // kM*384 floats  (34.6 MB)
  float* msda = qp   + (size_t)kM * 384;     // kM*256 floats  (23.0 MB)

  constexpr int NQ = kH * kL * kP * 3;       // 384
  const int tilesM = (kM + 15) / 16;

  // 1) img_p = img @ W_img + b_img
  {
    const int waves = tilesM * (256 / 64);
    gemm_bias_wmma_f32<256, kEMB><<<(waves + 7) / 8, 256, 0, stream>>>(
        img, W_img, b_img, imgp, kM);
  }
  // 2) qp = queries @ W_q + b_q
  {
    const int waves = tilesM * (NQ / 64);
    gemm_bias_wmma_f32<NQ, kEMB><<<(waves + 7) / 8, 256, 0, stream>>>(
        queries, W_q, b_q, qp, kM);
  }
  // 3) softmax over L*P logits + add reference points (in place on qp)
  softmax_prep<<<(kM * kH + 255) / 256, 256, 0, stream>>>(qp, refpts);
  // 4) deformable bilinear sampling -> msda (B,N,H,Cp)
  msda_sample<<<(kM * kH + 7) / 8, 256, 0, stream>>>(imgp, qp, msda);
  // 5) out = msda @ W_out + b_out
  {
    const int waves = tilesM * (256 / 64);
    gemm_bias_wmma_f32<256, 256><<<(waves + 7) / 8, 256, 0, stream>>>(
        msda, W_out, b_out, out, kM);
  }
}